// PackedHGTConv_86406152061290
// MI455X (gfx1250) — compile-verified
//
#include <hip/hip_runtime.h>
#include <hip/hip_bf16.h>
#include <math.h>

// ---------------------------------------------------------------------------
// Types for WMMA fragments (wave32, V_WMMA_F32_16X16X4_F32)
// ---------------------------------------------------------------------------
typedef __attribute__((ext_vector_type(2))) float v2f;
typedef __attribute__((ext_vector_type(8))) float v8f;

#define D_DIM 128
#define H_HEADS 8
#define DH_DIM 16

// ---------------------------------------------------------------------------
// Utility: fill buffer with a constant
// ---------------------------------------------------------------------------
__global__ void hgt_fillf(float* __restrict__ p, float val, size_t n) {
    size_t i = (size_t)blockIdx.x * blockDim.x + threadIdx.x;
    if (i < n) p[i] = val;
}

// ---------------------------------------------------------------------------
// Float atomic max via CAS (portable; scores are finite, init is -inf)
// ---------------------------------------------------------------------------
__device__ __forceinline__ void atomicMaxF(float* addr, float val) {
    int* ai = (int*)addr;
    int cur = __float_as_int(*addr);
    while (__int_as_float(cur) < val) {
        int prev = atomicCAS(ai, cur, __float_as_int(val));
        if (prev == cur) break;
        cur = prev;
    }
}

// ---------------------------------------------------------------------------
// WMMA GEMM:  C[n, j] = sum_d A[n,d] * W[j,d] + bias[j] (+ resid[n,j])
// A: [nrows,128] row-major, W: [128,128] row-major (out,in) -> implicit W^T.
//
// Block = 256 threads = 8 waves; block computes a 128-row x 128-col tile.
// W (64 KB) is staged once per block into LDS, pre-swizzled into WMMA
// fragment order:  lds_b[(kstep*8 + jtile)*32 + lane]  (float2 per lane).
// Per-lane LDS read stride = 8 B = 2 banks -> 32 lanes cover all 64 banks,
// conflict-free ds_load_b64 broadcasts across the 8 waves.
//
// Fragment layout (ISA 7.12.2, 32-bit, wave32):
//   A 16x4:  lane L<16 -> {A[M=L][k0], A[M=L][k0+1]}, lane L+16 -> k0+2,k0+3
//   B 4x16:  lane L<16 -> {B[k0][N=L], B[k0+1][N=L]}, lane L+16 -> k0+2,k0+3
//   B[k][n] = W[n][k]
//   C/D 16x16: lane L, vgpr i -> M = i + 8*(L>>4), N = L&15
// ---------------------------------------------------------------------------
__global__ void hgt_gemm_wmma(const float* __restrict__ A,
                              const float* __restrict__ W,
                              const float* __restrict__ bias,
                              const float* __restrict__ resid,
                              float* __restrict__ C,
                              int nrows) {
    __shared__ v2f lds_b[32 * 8 * 32];   // [kstep][jtile][lane], 64 KB

    // ---- stage W into LDS, coalesced reads, fragment-order writes ----
    for (int i = threadIdx.x; i < (D_DIM * D_DIM) / 2; i += 256) {
        const int fi  = i * 2;               // linear float index into W
        const int row = fi >> 7;             // W row (output channel)
        const int col = fi & 127;            // W col (input channel), even
        v2f w = *(const v2f*)(W + (size_t)row * D_DIM + col);
        const int ks   = col >> 2;                       // k-step
        const int lane = (row & 15) + ((col & 2) << 3);  // +16 for K hi-half
        const int jj   = row >> 4;                       // column tile
        lds_b[(ks * 8 + jj) * 32 + lane] = w;
    }
    __syncthreads();

    const int wave = threadIdx.x >> 5;
    const int lane = threadIdx.x & 31;
    const int row0 = (blockIdx.x * 8 + wave) * 16;
    if (row0 >= nrows) return;            // wave-uniform exit: EXEC stays all-ones

    const int lr = lane & 15;             // column-in-tile / row-in-tile index
    const int kh = (lane >> 4) << 1;      // K sub-offset: 0 or 2

    v8f acc[8];
    const v8f zero = {0.f, 0.f, 0.f, 0.f, 0.f, 0.f, 0.f, 0.f};
#pragma unroll
    for (int j = 0; j < 8; ++j) acc[j] = zero;

    const float* arow = A + (size_t)(row0 + lr) * D_DIM + kh;
    const v2f* bbase = lds_b + lane;

#pragma unroll 4
    for (int ks = 0; ks < 32; ++ks) {
        v2f a = *(const v2f*)(arow + ks * 4);     // streaming global A fragment
        v2f b[8];
#pragma unroll
        for (int j = 0; j < 8; ++j) b[j] = bbase[(ks * 8 + j) * 32];
#pragma unroll
        for (int j = 0; j < 8; ++j) {
            acc[j] = __builtin_amdgcn_wmma_f32_16x16x4_f32(
                /*neg_a=*/false, a, /*neg_b=*/false, b[j],
                /*c_mod=*/(short)0, acc[j],
                /*reuse_a=*/false, /*reuse_b=*/false);
        }
    }

    const int rsub = 8 * (lane >> 4);
#pragma unroll
    for (int j = 0; j < 8; ++j) {
        const int col = j * 16 + lr;
        const float bval = bias ? bias[col] : 0.f;
#pragma unroll
        for (int i = 0; i < 8; ++i) {
            const int r = row0 + rsub + i;
            float val = acc[j][i] + bval;
            if (resid) val += resid[(size_t)r * D_DIM + col];
            C[(size_t)r * D_DIM + col] = val;
        }
    }
}

// ---------------------------------------------------------------------------
// Compose per-relation key transform into projection weights:
//   Ck[h*16+kk][d] = sum_dd edgeW[rel,h,dd,kk] * Wk[h*16+dd][d]
//   cb[h*16+kk]    = sum_dd edgeW[rel,h,dd,kk] * bk[h*16+dd]
// edge_W flat index: ((rel*8 + h)*16 + dd)*16 + kk
// ---------------------------------------------------------------------------
__global__ void hgt_compose(const float* __restrict__ Wk,
                            const float* __restrict__ bk,
                            const float* __restrict__ edgeW,
                            int rel,
                            float* __restrict__ Ck,
                            float* __restrict__ cb) {
    int idx = blockIdx.x * blockDim.x + threadIdx.x;
    if (idx < D_DIM * D_DIM) {
        int o = idx / D_DIM, d = idx % D_DIM;
        int h = o >> 4, kk = o & 15;
        const float* ew = edgeW + ((size_t)rel * H_HEADS + h) * (DH_DIM * DH_DIM) + kk;
        float s = 0.f;
#pragma unroll
        for (int dd = 0; dd < DH_DIM; ++dd)
            s += ew[dd * DH_DIM] * Wk[(size_t)(h * 16 + dd) * D_DIM + d];
        Ck[(size_t)o * D_DIM + d] = s;
    } else if (idx < D_DIM * D_DIM + D_DIM) {
        int o = idx - D_DIM * D_DIM;
        int h = o >> 4, kk = o & 15;
        const float* ew = edgeW + ((size_t)rel * H_HEADS + h) * (DH_DIM * DH_DIM) + kk;
        float s = 0.f;
#pragma unroll
        for (int dd = 0; dd < DH_DIM; ++dd)
            s += ew[dd * DH_DIM] * bk[h * 16 + dd];
        cb[o] = s;
    }
}

// ---------------------------------------------------------------------------
// Pass A: one wave per edge. score[e,h] = q[dst]·kt[src] per head / sqrt(DH),
// store score, atomic-max into mx[dst,h], atomic count deg[dst].
// Lane l covers channels 4l..4l+3 (head = l>>2); 4-lane shuffle reduction.
// ---------------------------------------------------------------------------
__global__ void hgt_edge_scores(const float* __restrict__ q,
                                const float* __restrict__ kt,
                                const int* __restrict__ src,
                                const int* __restrict__ dst,
                                float* __restrict__ scoreBuf,
                                float* __restrict__ mx,
                                float* __restrict__ deg,
                                int nedges) {
    int gid = blockIdx.x * blockDim.x + threadIdx.x;
    int e = gid >> 5;
    int lane = threadIdx.x & 31;
    if (e >= nedges) return;
    int s = src[e], d = dst[e];
    float4 a = ((const float4*)(q + (size_t)d * D_DIM))[lane];
    float4 b = ((const float4*)(kt + (size_t)s * D_DIM))[lane];
    float p = a.x * b.x + a.y * b.y + a.z * b.z + a.w * b.w;
    p += __shfl_xor(p, 1, 32);
    p += __shfl_xor(p, 2, 32);
    if ((lane & 3) == 0) {
        int h = lane >> 2;
        float sc = p * 0.25f;                       // 1/sqrt(DH=16)
        scoreBuf[(size_t)e * H_HEADS + h] = sc;
        atomicMaxF(mx + (size_t)d * H_HEADS + h, sc);
    }
    if (lane == 0) atomicAdd(deg + d, 1.0f);
}

// ---------------------------------------------------------------------------
// Pass B: ex = exp(score - mx[dst]); overwrite scoreBuf; atomicAdd into sm.
// One thread per (edge, head).
// ---------------------------------------------------------------------------
__global__ void hgt_edge_expsum(const int* __restrict__ dst,
                                float* __restrict__ scoreBuf,
                                const float* __restrict__ mx,
                                float* __restrict__ sm,
                                int nedges) {
    int i = blockIdx.x * blockDim.x + threadIdx.x;
    if (i >= nedges * H_HEADS) return;
    int e = i >> 3, h = i & 7;
    int d = dst[e];
    float ex = expf(scoreBuf[i] - mx[(size_t)d * H_HEADS + h]);
    scoreBuf[i] = ex;
    atomicAdd(sm + (size_t)d * H_HEADS + h, ex);
}

// ---------------------------------------------------------------------------
// Pass C: attn = ex / max(sm,eps); msg[dst] += attn * v[src]. Wave per edge.
// ---------------------------------------------------------------------------
__global__ void hgt_edge_scatter(const float* __restrict__ v,
                                 const int* __restrict__ src,
                                 const int* __restrict__ dst,
                                 const float* __restrict__ exBuf,
                                 const float* __restrict__ sm,
                                 float* __restrict__ msg,
                                 int nedges) {
    int gid = blockIdx.x * blockDim.x + threadIdx.x;
    int e = gid >> 5;
    int lane = threadIdx.x & 31;
    if (e >= nedges) return;
    int s = src[e], d = dst[e];
    int h = lane >> 2;
    float attn = exBuf[(size_t)e * H_HEADS + h] /
                 fmaxf(sm[(size_t)d * H_HEADS + h], 1e-8f);
    float4 val = ((const float4*)(v + (size_t)s * D_DIM))[lane];
    float* mp = msg + (size_t)d * D_DIM + lane * 4;
    atomicAdd(mp + 0, attn * val.x);
    atomicAdd(mp + 1, attn * val.y);
    atomicAdd(mp + 2, attn * val.z);
    atomicAdd(mp + 3, attn * val.w);
}

// ---------------------------------------------------------------------------
// Gate over relations (degree-scaled softmax) + GeLU (tanh approx, as JAX).
// One thread per (node, channel).
// ---------------------------------------------------------------------------
__global__ void hgt_combine_gelu(const float* __restrict__ m0,
                                 const float* __restrict__ m1,
                                 const float* __restrict__ d0,
                                 const float* __restrict__ d1,
                                 const float* __restrict__ gate,
                                 float* __restrict__ g,
                                 int n) {
    int i = blockIdx.x * blockDim.x + threadIdx.x;
    if (i >= n * D_DIM) return;
    int node = i >> 7;
    float dg0 = d0[node], dg1 = d1[node];
    float l0 = gate[0] + log1pf(dg0);
    float l1 = gate[1] + log1pf(dg1);
    const float NEG = -1e30f;
    float a0 = (dg0 > 0.f) ? l0 : NEG;
    float a1 = (dg1 > 0.f) ? l1 : NEG;
    float mxl = fmaxf(a0, a1);
    float e0 = expf(a0 - mxl), e1 = expf(a1 - mxl);
    float inv = 1.f / (e0 + e1);
    float hm = (dg0 + dg1 > 0.f) ? (e0 * inv * m0[i] + e1 * inv * m1[i]) : 0.f;
    // jax.nn.gelu default: tanh approximation
    float x = hm;
    float t = tanhf(0.7978845608028654f * (x + 0.044715f * x * x * x));
    g[i] = 0.5f * x * (1.f + t);
}

// ---------------------------------------------------------------------------
// LayerNorm over 128 channels: one wave per node, 4 channels/lane.
// ---------------------------------------------------------------------------
__global__ void hgt_layernorm(const float* __restrict__ res,
                              const float* __restrict__ scale,
                              const float* __restrict__ bias,
                              float* __restrict__ out,
                              int n) {
    int gid = blockIdx.x * blockDim.x + threadIdx.x;
    int node = gid >> 5;
    int lane = threadIdx.x & 31;
    if (node >= n) return;
    float4 x = ((const float4*)(res + (size_t)node * D_DIM))[lane];
    float s = x.x + x.y + x.z + x.w;
#pragma unroll
    for (int o = 1; o < 32; o <<= 1) s += __shfl_xor(s, o, 32);
    float mu = s * (1.f / 128.f);
    float dx = x.x - mu, dy = x.y - mu, dz = x.z - mu, dw = x.w - mu;
    float sq = dx * dx + dy * dy + dz * dz + dw * dw;
#pragma unroll
    for (int o = 1; o < 32; o <<= 1) sq += __shfl_xor(sq, o, 32);
    float rstd = rsqrtf(sq * (1.f / 128.f) + 1e-5f);
    float4 sc = ((const float4*)scale)[lane];
    float4 bi = ((const float4*)bias)[lane];
    float4 o4;
    o4.x = dx * rstd * sc.x + bi.x;
    o4.y = dy * rstd * sc.y + bi.y;
    o4.z = dz * rstd * sc.z + bi.z;
    o4.w = dw * rstd * sc.w + bi.w;
    ((float4*)(out + (size_t)node * D_DIM))[lane] = o4;
}

// ---------------------------------------------------------------------------
// Orchestration
// ---------------------------------------------------------------------------
extern "C" void kernel_launch(void* const* d_in, const int* in_sizes, int n_in,
                              void* d_out, int out_size, void* d_ws, size_t ws_size,
                              hipStream_t stream) {
    const float* h     = (const float*)d_in[0];
    const float* Wq    = (const float*)d_in[1];
    const float* bq    = (const float*)d_in[2];
    const float* Wk    = (const float*)d_in[3];
    const float* bk    = (const float*)d_in[4];
    const float* Wv    = (const float*)d_in[5];
    const float* bv    = (const float*)d_in[6];
    const float* edgeW = (const float*)d_in[7];
    const float* gate  = (const float*)d_in[8];
    const float* Wo    = (const float*)d_in[9];
    const float* bo    = (const float*)d_in[10];
    const float* lns   = (const float*)d_in[11];
    const float* lnb   = (const float*)d_in[12];
    const int* srcs[2] = {(const int*)d_in[13], (const int*)d_in[15]};
    const int* dsts[2] = {(const int*)d_in[14], (const int*)d_in[16]};

    const int N = in_sizes[0] / D_DIM;
    const int E = in_sizes[13];

    // ---- workspace layout (floats) ----
    float* ws = (float*)d_ws;
    const size_t nd = (size_t)N * D_DIM;
    float* q        = ws;                    // nd
    float* vbuf     = q + nd;                // nd
    float* kt       = vbuf + nd;             // nd   (reused per relation; later = g)
    float* m0       = kt + nd;               // nd
    float* m1       = m0 + nd;               // nd
    float* scoreBuf = m1 + nd;               // E*8  (reused across relations)
    float* mx       = scoreBuf + (size_t)E * H_HEADS;   // N*8
    float* sm       = mx + (size_t)N * H_HEADS;         // N*8
    float* deg0     = sm + (size_t)N * H_HEADS;         // N
    float* deg1     = deg0 + N;                          // N
    float* Ck       = deg1 + N;                          // 128*128
    float* cbk      = Ck + D_DIM * D_DIM;                // 128
    float* g        = kt;   // alias: kt dead after edge passes
    float* res      = q;    // alias: q dead after edge passes
    float* msgs[2]  = {m0, m1};
    float* degs[2]  = {deg0, deg1};

    const int gemmBlocks  = (N / 16 + 7) / 8;            // 8 waves (16 rows each)/block
    const int waveBlocks  = (E + 7) / 8;                 // 1 wave per edge, 8/block
    const int ehBlocks    = (E * H_HEADS + 255) / 256;
    const int ncBlocks    = (N * D_DIM + 255) / 256;
    const int lnBlocks    = (N + 7) / 8;

    // q, v projections (fp32 WMMA)
    hgt_gemm_wmma<<<gemmBlocks, 256, 0, stream>>>(h, Wq, bq, nullptr, q, N);
    hgt_gemm_wmma<<<gemmBlocks, 256, 0, stream>>>(h, Wv, bv, nullptr, vbuf, N);

    for (int rel = 0; rel < 2; ++rel) {
        // composed key-transform projection:  kt = h @ Ck^T + cbk
        hgt_compose<<<(D_DIM * D_DIM + D_DIM + 255) / 256, 256, 0, stream>>>(
            Wk, bk, edgeW, rel, Ck, cbk);
        hgt_gemm_wmma<<<gemmBlocks, 256, 0, stream>>>(h, Ck, cbk, nullptr, kt, N);

        // per-relation accumulator init
        hgt_fillf<<<((size_t)N * H_HEADS + 255) / 256, 256, 0, stream>>>(
            mx, -INFINITY, (size_t)N * H_HEADS);
        hgt_fillf<<<((size_t)N * H_HEADS + 255) / 256, 256, 0, stream>>>(
            sm, 0.f, (size_t)N * H_HEADS);
        hgt_fillf<<<(N + 255) / 256, 256, 0, stream>>>(degs[rel], 0.f, (size_t)N);
        hgt_fillf<<<ncBlocks, 256, 0, stream>>>(msgs[rel], 0.f, nd);

        // segmented softmax + scatter
        hgt_edge_scores<<<waveBlocks, 256, 0, stream>>>(
            q, kt, srcs[rel], dsts[rel], scoreBuf, mx, degs[rel], E);
        hgt_edge_expsum<<<ehBlocks, 256, 0, stream>>>(
            dsts[rel], scoreBuf, mx, sm, E);
        hgt_edge_scatter<<<waveBlocks, 256, 0, stream>>>(
            vbuf, srcs[rel], dsts[rel], scoreBuf, sm, msgs[rel], E);
    }

    // gate + GeLU, output projection with fused bias+residual, LayerNorm
    hgt_combine_gelu<<<ncBlocks, 256, 0, stream>>>(m0, m1, deg0, deg1, gate, g, N);
    hgt_gemm_wmma<<<gemmBlocks, 256, 0, stream>>>(g, Wo, bo, h, res, N);
    hgt_layernorm<<<lnBlocks, 256, 0, stream>>>(res, lns, lnb, (float*)d_out, N);
}